// EqualizedConvTranspose_34359738368711
// MI455X (gfx1250) — compile-verified
//
#include <hip/hip_runtime.h>
#include <math.h>

typedef __attribute__((ext_vector_type(2))) float v2f;
typedef __attribute__((ext_vector_type(8))) float v8f;

#define CIN 16
#define COUT 32
#define TILES_PER_WAVE 10
#define WAVES_PER_BLOCK 8

// ---------------------------------------------------------------------------
// Kernel 1: out[r][c] = bias[c] for all rows (vectorized float4 stores).
// ---------------------------------------------------------------------------
__global__ void __launch_bounds__(256) bias_init_kernel(const float* __restrict__ bias,
                                                        float* __restrict__ out, int n4) {
  int i = blockIdx.x * blockDim.x + threadIdx.x;
  if (i >= n4) return;
  int c = (i & 7) << 2;                       // 32 channels = 8 float4 per row
  float4 v = make_float4(bias[c], bias[c + 1], bias[c + 2], bias[c + 3]);
  reinterpret_cast<float4*>(out)[i] = v;
}

// ---------------------------------------------------------------------------
// Kernel 2: gather 16 input rows, WMMA f32 16x16x4 chained over K=16
// (two N-halves for COUT=32), scatter-add with f32 atomics.
// blockIdx.y = kernel offset k (no integer division); one wave owns a fixed k
// and processes TILES_PER_WAVE tiles of 16 pairs, so B (= scaled kernel[k])
// is loaded into registers exactly once per wave.
// ---------------------------------------------------------------------------
__global__ void __launch_bounds__(256) scatter_conv_wmma_kernel(
    const float* __restrict__ x, const float* __restrict__ kern,
    const int* __restrict__ in_map, const int* __restrict__ out_map,
    float* __restrict__ out, int M, int tilesPerK, float scale) {
  const int k    = blockIdx.y;                                  // kernel offset
  const int slot = blockIdx.x * WAVES_PER_BLOCK + (int)(threadIdx.x >> 5);

  const int lane  = threadIdx.x & 31;
  const int ncol  = lane & 15;               // A: row M ; B/C: column N
  const int khalf = lane >> 4;               // 0 -> K rows {0,1}+4s ; 1 -> {2,3}+4s

  // ---- B = kernel[k] (16x32), equalized scale folded in, in registers ----
  const float* kb = kern + (long long)k * (CIN * COUT);
  v2f b[4][2];
#pragma unroll
  for (int s = 0; s < 4; ++s) {
    const int kk = 4 * s + 2 * khalf;
#pragma unroll
    for (int h = 0; h < 2; ++h) {
      const int n = h * 16 + ncol;
      v2f bb;
      bb.x = kb[kk * COUT + n] * scale;
      bb.y = kb[(kk + 1) * COUT + n] * scale;
      b[s][h] = bb;
    }
  }

  const int* imap = in_map + (long long)k * M;
  const int* omap = out_map + (long long)k * M;

  for (int t = 0; t < TILES_PER_WAVE; ++t) {
    const int tile = slot * TILES_PER_WAVE + t;
    if (tile >= tilesPerK) break;            // uniform per wave -> EXEC stays all-1s
    const int m_base = tile * 16;

    // ---- gather A: 16 rows x 16 channels; each lane: 4x float2 loads ----
    const int arow = imap[m_base + ncol];
    const float* xr = x + (long long)arow * CIN + 2 * khalf;
    v2f a0 = *(const v2f*)(xr + 0);
    v2f a1 = *(const v2f*)(xr + 4);
    v2f a2 = *(const v2f*)(xr + 8);
    v2f a3 = *(const v2f*)(xr + 12);

    v8f c0 = {};  // N = 0..15
    v8f c1 = {};  // N = 16..31
    c0 = __builtin_amdgcn_wmma_f32_16x16x4_f32(false, a0, false, b[0][0], (short)0, c0, false, false);
    c0 = __builtin_amdgcn_wmma_f32_16x16x4_f32(false, a1, false, b[1][0], (short)0, c0, false, false);
    c0 = __builtin_amdgcn_wmma_f32_16x16x4_f32(false, a2, false, b[2][0], (short)0, c0, false, false);
    c0 = __builtin_amdgcn_wmma_f32_16x16x4_f32(false, a3, false, b[3][0], (short)0, c0, false, false);
    c1 = __builtin_amdgcn_wmma_f32_16x16x4_f32(false, a0, false, b[0][1], (short)0, c1, false, false);
    c1 = __builtin_amdgcn_wmma_f32_16x16x4_f32(false, a1, false, b[1][1], (short)0, c1, false, false);
    c1 = __builtin_amdgcn_wmma_f32_16x16x4_f32(false, a2, false, b[2][1], (short)0, c1, false, false);
    c1 = __builtin_amdgcn_wmma_f32_16x16x4_f32(false, a3, false, b[3][1], (short)0, c1, false, false);

    // ---- scatter-add: C VGPR j holds row M = j + 8*khalf, col = ncol ----
    // 8 consecutive out rows -> two 16B-aligned b128 index loads.
    const int4* op4 = (const int4*)(omap + m_base + khalf * 8);
    const int4 o0 = op4[0];
    const int4 o1 = op4[1];
    const int orow[8] = {o0.x, o0.y, o0.z, o0.w, o1.x, o1.y, o1.z, o1.w};
#pragma unroll
    for (int j = 0; j < 8; ++j) {
      float* op = out + (long long)orow[j] * COUT + ncol;
      atomicAdd(op, c0[j]);        // global_atomic_add_f32 (no-return), lands in L2
      atomicAdd(op + 16, c1[j]);
    }
  }
}

// ---------------------------------------------------------------------------
// Inputs (setup_inputs order): x[f32], kernel[f32], bias[f32], in_map[i32],
// out_map[i32], n_out[i32 scalar]. Output: out[f32] (n_out x 32).
// ---------------------------------------------------------------------------
extern "C" void kernel_launch(void* const* d_in, const int* in_sizes, int n_in,
                              void* d_out, int out_size, void* d_ws, size_t ws_size,
                              hipStream_t stream) {
  const float* x       = (const float*)d_in[0];
  const float* kern    = (const float*)d_in[1];
  const float* bias    = (const float*)d_in[2];
  const int*   in_map  = (const int*)d_in[3];
  const int*   out_map = (const int*)d_in[4];
  float*       out     = (float*)d_out;

  const int nK = in_sizes[1] / (CIN * COUT);        // 27
  const int M  = in_sizes[3] / nK;                  // 200000 (multiple of 16)
  const float scale = sqrtf(2.0f / (float)(nK * CIN));

  // 1) broadcast bias into out
  const int n4 = out_size / 4;
  bias_init_kernel<<<(n4 + 255) / 256, 256, 0, stream>>>(bias, out, n4);

  // 2) gather -> WMMA -> atomic scatter (2D grid: y = kernel offset)
  const int tilesPerK = M / 16;                                     // 12500
  const int wavesPerK = (tilesPerK + TILES_PER_WAVE - 1) / TILES_PER_WAVE;
  const int blocksX   = (wavesPerK + WAVES_PER_BLOCK - 1) / WAVES_PER_BLOCK;
  dim3 grid(blocksX, nK);
  scatter_conv_wmma_kernel<<<grid, 32 * WAVES_PER_BLOCK, 0, stream>>>(
      x, kern, in_map, out_map, out, M, tilesPerK, scale);
}